// CrossAttnBlock_7619271983717
// MI455X (gfx1250) — compile-verified
//
#include <hip/hip_runtime.h>
#include <hip/hip_bf16.h>

// Problem constants from the reference
#define BATCH 8
#define CCH   256      // channels C
#define CTX   512      // context dim CC
#define HW    2304     // 48*48

typedef float v2f __attribute__((ext_vector_type(2)));
typedef float v4f __attribute__((ext_vector_type(4)));
typedef float v8f __attribute__((ext_vector_type(8)));

// -----------------------------------------------------------------------------
// Kernel 1: proj[b][o] = b3[o] + sum_c w3[o][c] * (b2[c] + sum_cc w2[c][cc]*context[b][cc])
// One workgroup, 16 waves (512 threads). Each wave owns a 16-wide N tile.
// Both GEMMs done with V_WMMA_F32_16X16X4_F32 (fp32-native CDNA5 WMMA),
// M padded from 8 -> 16 with zero rows.
//
// A-fragment (16x4 f32, ISA 7.12.2): lane%16 = M row; lane/16 selects K pair:
//   a.x = A[M][k0 + 2*(lane>>4)], a.y = A[M][k0 + 2*(lane>>4) + 1]
// B-fragment (4x16 f32): same striping with N = lane%16:
//   b.x = B[k0 + 2*(lane>>4)][N], b.y = B[k0 + 2*(lane>>4)+1][N]
// C/D (16x16 f32, 8 VGPRs): d[r] = D[M = r + 8*(lane>>4)][N = lane%16]
// -----------------------------------------------------------------------------
__global__ void __launch_bounds__(512)
crossattn_proj_wmma(const float* __restrict__ context, // [8,512]
                    const float* __restrict__ w2,      // [256,512]
                    const float* __restrict__ b2,      // [256]
                    const float* __restrict__ w3,      // [256,256]
                    const float* __restrict__ b3,      // [256]
                    float* __restrict__ proj)          // [8,256] in d_ws
{
    __shared__ float vbuf[BATCH][CCH];      // v = context @ w2^T + b2

    const int lane  = threadIdx.x & 31;
    const int wave  = threadIdx.x >> 5;     // 0..15 -> N tile
    const int row   = lane & 15;            // M (for A) / N (for B) within tile
    const int koff  = (lane >> 4) * 2;      // K sub-offset inside 4-wide K step
    const int n     = wave * 16 + row;      // global N column this lane feeds B with
    const float msk = (row < BATCH) ? 1.0f : 0.0f;  // zero-pad rows 8..15
    const int rsrc  = row & (BATCH - 1);    // safe row index for padded loads

    // ---- Stage 1: v[16 x 256] = Apad(context)[16 x 512] * w2^T[512 x 256] ----
    v8f acc = {};
    for (int k0 = 0; k0 < CTX; k0 += 4) {
        const int ka = k0 + koff;
        v2f a, b;
        a.x = msk * context[rsrc * CTX + ka];
        a.y = msk * context[rsrc * CTX + ka + 1];
        b.x = w2[n * CTX + ka];             // B[k][n] = w2[n][k]
        b.y = w2[n * CTX + ka + 1];
        acc = __builtin_amdgcn_wmma_f32_16x16x4_f32(
                  false, a, false, b, (short)0, acc, false, false);
    }
    // lanes 0..15 hold M = r (r = 0..7 are the valid batch rows)
    if (lane < 16) {
        #pragma unroll
        for (int r = 0; r < BATCH; ++r)
            vbuf[r][wave * 16 + lane] = acc[r] + b2[wave * 16 + lane];
    }
    __syncthreads();

    // ---- Stage 2: proj[16 x 256] = Vpad[16 x 256] * w3^T[256 x 256] ----
    v8f acc2 = {};
    for (int k0 = 0; k0 < CCH; k0 += 4) {
        const int ka = k0 + koff;
        v2f a, b;
        a.x = msk * vbuf[rsrc][ka];
        a.y = msk * vbuf[rsrc][ka + 1];
        b.x = w3[n * CCH + ka];             // B[k][n] = w3[n][k]
        b.y = w3[n * CCH + ka + 1];
        acc2 = __builtin_amdgcn_wmma_f32_16x16x4_f32(
                   false, a, false, b, (short)0, acc2, false, false);
    }
    if (lane < 16) {
        #pragma unroll
        for (int r = 0; r < BATCH; ++r)
            proj[r * CCH + wave * 16 + lane] = acc2[r] + b3[wave * 16 + lane];
    }
}

// -----------------------------------------------------------------------------
// Kernel 2: out[b,c,h,w] = x[b,c,h,w] + proj[b,c]
// One block per (b,c) row (2048 blocks). proj[bc] is block-uniform -> scalar
// load. Body streams 2304 floats as 576 x 16B with nontemporal hints
// (single-touch data; keep it out of L2). Pure HBM-bound: ~37.7 MB total.
// -----------------------------------------------------------------------------
__global__ void __launch_bounds__(192)
crossattn_residual_add(const float* __restrict__ x,
                       const float* __restrict__ proj,
                       float* __restrict__ out)
{
    const int bc = blockIdx.x;              // b*256 + c, 0..2047
    const float p = proj[bc];               // uniform across block

    const v4f* __restrict__ x4 = (const v4f*)(x + (size_t)bc * HW);
    v4f* __restrict__ o4 = (v4f*)(out + (size_t)bc * HW);

    #pragma unroll
    for (int it = 0; it < 3; ++it) {        // 3 * 192 = 576 vec4 exactly
        const int i = it * 192 + threadIdx.x;
        v4f v = __builtin_nontemporal_load(&x4[i]);
        v.x += p; v.y += p; v.z += p; v.w += p;
        __builtin_nontemporal_store(v, &o4[i]);
    }
}

extern "C" void kernel_launch(void* const* d_in, const int* in_sizes, int n_in,
                              void* d_out, int out_size, void* d_ws, size_t ws_size,
                              hipStream_t stream) {
    // setup_inputs() order:
    // 0:x 1:context 2:gn_scale 3:gn_bias 4:w0 5:b0 6:w1 7:b1 8:w2 9:b2 10:w3 11:b3
    const float* x       = (const float*)d_in[0];
    const float* context = (const float*)d_in[1];
    const float* w2      = (const float*)d_in[8];
    const float* b2      = (const float*)d_in[9];
    const float* w3      = (const float*)d_in[10];
    const float* b3      = (const float*)d_in[11];
    float* out  = (float*)d_out;
    float* proj = (float*)d_ws;             // 8*256 floats = 8 KB scratch

    // Tiny double-GEMM via fp32 WMMA: one workgroup, 16 waves.
    crossattn_proj_wmma<<<1, 512, 0, stream>>>(context, w2, b2, w3, b3, proj);

    // Streaming residual broadcast-add: 2048 (b,c) rows.
    crossattn_residual_add<<<BATCH * CCH, 192, 0, stream>>>(x, proj, out);
}